// S4dplr_83734682403186
// MI455X (gfx1250) — compile-verified
//
#include <hip/hip_runtime.h>
#include <math.h>

// Problem sizes (fixed by the reference)
#define BATCH 8
#define SEQ   4096          // L = 64*64 -> four-step FFT with DFT-64 GEMMs
#define DDIM  256
#define HDIM  64
#define PI_F  3.14159265358979323846f

typedef __attribute__((ext_vector_type(2))) float v2f;   // f32 WMMA A/B frag (16x4 / 4x16)
typedef __attribute__((ext_vector_type(8))) float v8f;   // f32 WMMA C/D frag (16x16)

// D = A(16x4) * B(4x16) + C, full f32 — V_WMMA_F32_16X16X4_F32
__device__ __forceinline__ v8f wmma4(v2f a, v2f b, v8f c) {
  return __builtin_amdgcn_wmma_f32_16x16x4_f32(false, a, false, b, (short)0, c,
                                               false, false);
}

// ---------------------------------------------------------------------------
// Kernel 1: at_roots[d,l] via Woodbury/Cauchy dot products (pure VALU, tiny)
// ---------------------------------------------------------------------------
__global__ __launch_bounds__(256) void cauchy_kernel(
    const float2* __restrict__ lam, const float2* __restrict__ P,
    const float2* __restrict__ Bc,  const float2* __restrict__ Cc,
    const float*  __restrict__ log_delta, float2* __restrict__ atr)
{
  int idx = blockIdx.x * 256 + threadIdx.x;   // over D*L
  int d = idx >> 12;                          // /4096
  int l = idx & (SEQ - 1);

  float delta = __expf(log_delta[d]);
  float zs, zc;
  __sincosf((2.0f * PI_F / SEQ) * (float)l, &zs, &zc);   // z = exp(2*pi*i*l/L)

  // coef = 2/(1+z);  g = (2*(1-z)/(1+z)) / delta
  float pr = 1.f + zc, pim = zs;
  float inv = 1.f / (pr * pr + pim * pim);
  float coefr =  2.f * pr * inv, coefi = -2.f * pim * inv;
  float mr = 1.f - zc, mi = -zs;
  float gr = 2.f * (mr * pr + mi * pim) * inv / delta;
  float gi = 2.f * (mi * pr - mr * pim) * inv / delta;

  float k00r = 0, k00i = 0, k01r = 0, k01i = 0;
  float k10r = 0, k10i = 0, k11r = 0, k11i = 0;
#pragma unroll 4
  for (int h = 0; h < HDIM; ++h) {
    float2 la = lam[h];
    float2 pp = P[h];
    float2 cb = Cc[d * HDIM + h];
    float2 bb = Bc[d * HDIM + h];
    // cauchy = 1 / (g - lam_h)
    float wr = gr - la.x, wi = gi - la.y;
    float iw = 1.f / (wr * wr + wi * wi);
    float cr = wr * iw, ci = -wi * iw;
    // weights: C*B, C*P, conj(P)*B, |P|^2
    float w00r = cb.x * bb.x - cb.y * bb.y, w00i = cb.x * bb.y + cb.y * bb.x;
    float w01r = cb.x * pp.x - cb.y * pp.y, w01i = cb.x * pp.y + cb.y * pp.x;
    float w10r = pp.x * bb.x + pp.y * bb.y, w10i = pp.x * bb.y - pp.y * bb.x;
    float w11  = pp.x * pp.x + pp.y * pp.y;
    k00r += w00r * cr - w00i * ci;  k00i += w00r * ci + w00i * cr;
    k01r += w01r * cr - w01i * ci;  k01i += w01r * ci + w01i * cr;
    k10r += w10r * cr - w10i * ci;  k10i += w10r * ci + w10i * cr;
    k11r += w11 * cr;               k11i += w11 * ci;
  }
  // at = coef * (k00 - k01*k10/(1+k11))
  float d1r = 1.f + k11r, d1i = k11i;
  float dinv = 1.f / (d1r * d1r + d1i * d1i);
  float nr = k01r * k10r - k01i * k10i, ni = k01r * k10i + k01i * k10r;
  float qr = (nr * d1r + ni * d1i) * dinv, qi = (ni * d1r - nr * d1i) * dinv;
  float wr_ = k00r - qr, wi_ = k00i - qi;
  atr[idx] = make_float2(coefr * wr_ - coefi * wi_,
                         coefr * wi_ + coefi * wr_);
}

// ---------------------------------------------------------------------------
// Kernel 2: Kf = fft(ifft(at_roots).real) done EXACTLY via Hermitian
// symmetrization, in place:  Kf[l] = 0.5*(a[l] + conj(a[(L-l)%L]))
// ---------------------------------------------------------------------------
__global__ __launch_bounds__(256) void symm_kernel(float2* __restrict__ a)
{
  int idx = blockIdx.x * 256 + threadIdx.x;   // over D * 2049 (pairs)
  if (idx >= DDIM * 2049) return;
  int d = idx / 2049, l = idx % 2049;
  int lm = (SEQ - l) & (SEQ - 1);
  float2* row = a + (size_t)d * SEQ;
  float2 u = row[l], v = row[lm];
  row[l]  = make_float2(0.5f * (u.x + v.x), 0.5f * (u.y - v.y));
  row[lm] = make_float2(0.5f * (v.x + u.x), 0.5f * (v.y - u.y));
}

// ---------------------------------------------------------------------------
// Kernel 3: transpose x (B,L,D) -> xt (B,D,L) so each FFT row is contiguous
// ---------------------------------------------------------------------------
__global__ __launch_bounds__(256) void transpose_in(const float* __restrict__ x,
                                                    float* __restrict__ xt)
{
  __shared__ float tile[32][33];
  int b = blockIdx.z, l0 = blockIdx.x * 32, d0 = blockIdx.y * 32;
  int tx = threadIdx.x, ty = threadIdx.y;   // (32,8)
#pragma unroll
  for (int r = 0; r < 4; ++r) {
    int l = l0 + ty + r * 8;
    tile[ty + r * 8][tx] = x[((size_t)b * SEQ + l) * DDIM + d0 + tx];
  }
  __syncthreads();
#pragma unroll
  for (int r = 0; r < 4; ++r) {
    int d = d0 + ty + r * 8;
    xt[((size_t)b * DDIM + d) * SEQ + l0 + tx] = tile[tx][ty + r * 8];
  }
}

// ---------------------------------------------------------------------------
// Kernel 4: fused forward-FFT * Kf * inverse-FFT per (b,d) row via WMMA.
// 4096-pt FFT = 64x64 four-step; each DFT-64 = GEMM with the constant W64
// matrix (only 64 distinct values -> 512B LDS LUT). 12 real 64^3 GEMMs per
// block; one workgroup (8 wave32) per row, 2 output tiles per wave.
// ---------------------------------------------------------------------------
__global__ __launch_bounds__(256) void fftconv_kernel(
    const float*  __restrict__ xt,   // (B, D, L)
    const float2* __restrict__ Kf,   // (D, L)
    float*        __restrict__ yt)   // (B, D, L)
{
  __shared__ float  sA[2 * 64 * 64];   // 64x64 complex, interleaved re/im (32KB)
  __shared__ float2 sWtab[64];         // W64 twiddle LUT: exp(-2*pi*i*j/64)

  const int tid  = threadIdx.x;
  const int lane = tid & 31;
  const int wv   = tid >> 5;
  const int bd   = blockIdx.x;             // b*DDIM + d
  const int d    = bd & (DDIM - 1);
  const float*  xrow  = xt + (size_t)bd * SEQ;
  float*        yrow  = yt + (size_t)bd * SEQ;
  const float2* kfrow = Kf + (size_t)d * SEQ;

  if (tid < 64) {
    float s, c;
    __sincosf(-2.f * PI_F * (float)tid / 64.f, &s, &c);
    sWtab[tid] = make_float2(c, s);
  }
  __syncthreads();

  const int r15   = lane & 15;           // row/col within 16
  const int khalf = (lane >> 4) << 1;    // K sub-offset: 0 or 2 (ISA A/B layout)
  const int rbofs = (lane >> 4) << 3;    // C/D row offset: 0 or 8 (ISA C/D layout)

  // --- WMMA fragment loaders (ISA 7.12.2 layouts) ---
  auto ldA_W = [&](int row0, int k0, float isign, v2f& ar, v2f& ai) {
    int r = row0 + r15, kk = k0 + khalf;
    float2 w0 = sWtab[(r * kk) & 63];
    float2 w1 = sWtab[(r * (kk + 1)) & 63];
    ar.x = w0.x; ar.y = w1.x; ai.x = isign * w0.y; ai.y = isign * w1.y;
  };
  auto ldB_W = [&](int k0, int col0, float isign, v2f& br, v2f& bi) {
    int c = col0 + r15, kk = k0 + khalf;
    float2 w0 = sWtab[(kk * c) & 63];
    float2 w1 = sWtab[((kk + 1) * c) & 63];
    br.x = w0.x; br.y = w1.x; bi.x = isign * w0.y; bi.y = isign * w1.y;
  };
  auto ldA_c = [&](int comp, int row0, int k0) -> v2f {
    int r = row0 + r15, kk = k0 + khalf;
    v2f a;
    a.x = sA[(r * 64 + kk) * 2 + comp];
    a.y = sA[(r * 64 + kk + 1) * 2 + comp];
    return a;
  };
  auto ldB_c = [&](int comp, int k0, int col0) -> v2f {
    int c = col0 + r15, kk = k0 + khalf;
    v2f b;
    b.x = sA[(kk * 64 + c) * 2 + comp];
    b.y = sA[((kk + 1) * 64 + c) * 2 + comp];
    return b;
  };
  auto ldB_g = [&](int k0, int col0) -> v2f {   // real input straight from global
    int c = col0 + r15, kk = k0 + khalf;
    v2f b;
    b.x = xrow[kk * 64 + c];
    b.y = xrow[(kk + 1) * 64 + c];
    return b;
  };

  // ========== forward stage 1:  A = W64 @ M   (M real -> 2 real GEMMs) =====
  v8f f1r[2], f1i[2];
#pragma unroll
  for (int t = 0; t < 2; ++t) {
    int tile = wv * 2 + t, tm = (tile >> 2) * 16, tn = (tile & 3) * 16;
    v8f ar = {}, ai = {};
#pragma unroll
    for (int k0 = 0; k0 < 64; k0 += 4) {
      v2f wr, wi; ldA_W(tm, k0, 1.f, wr, wi);
      v2f bm = ldB_g(k0, tn);
      ar = wmma4(wr, bm, ar);
      ai = wmma4(wi, bm, ai);
    }
    f1r[t] = ar; f1i[t] = ai;
  }
#pragma unroll
  for (int t = 0; t < 2; ++t) {
    int tile = wv * 2 + t, tm = (tile >> 2) * 16, tn = (tile & 3) * 16;
    int colg = tn + r15, rbase = tm + rbofs;
#pragma unroll
    for (int j = 0; j < 8; ++j) {
      int rowg = rbase + j;
      sA[(rowg * 64 + colg) * 2 + 0] = f1r[t][j];
      sA[(rowg * 64 + colg) * 2 + 1] = f1i[t][j];
    }
  }
  __syncthreads();

  // forward twiddle: A[k1][n2] *= exp(-2*pi*i*k1*n2/4096)
#pragma unroll
  for (int e = 0; e < 16; ++e) {
    int idx = tid * 16 + e;
    int k1 = idx >> 6, n2 = idx & 63;
    float s, c;
    __sincosf(-2.f * PI_F * (float)(k1 * n2) * (1.f / 4096.f), &s, &c);
    float re = sA[idx * 2], im = sA[idx * 2 + 1];
    sA[idx * 2]     = re * c - im * s;
    sA[idx * 2 + 1] = re * s + im * c;
  }
  __syncthreads();

  // ========== forward stage 2:  Y = A @ W64 (4 GEMMs), then * Kf / L =======
  float zr_[2][8], zi_[2][8];
#pragma unroll
  for (int t = 0; t < 2; ++t) {
    int tile = wv * 2 + t, tm = (tile >> 2) * 16, tn = (tile & 3) * 16;
    v8f yr = {}, yi = {};
#pragma unroll
    for (int k0 = 0; k0 < 64; k0 += 4) {
      v2f ar = ldA_c(0, tm, k0);
      v2f ai = ldA_c(1, tm, k0);
      v2f br, bi; ldB_W(k0, tn, 1.f, br, bi);
      yr = wmma4(ar, br, yr);
      yi = wmma4(ar, bi, yi);
      yi = wmma4(ai, br, yi);
      v2f nbi; nbi.x = -bi.x; nbi.y = -bi.y;    // f32 WMMA NEG is C-only
      yr = wmma4(ai, nbi, yr);
    }
    int colg = tn + r15, rbase = tm + rbofs;
#pragma unroll
    for (int j = 0; j < 8; ++j) {
      int rowg = rbase + j;                 // k1
      int k = colg * 64 + rowg;             // natural frequency index
      float2 kf = kfrow[k];
      zr_[t][j] = (yr[j] * kf.x - yi[j] * kf.y) * (1.f / 4096.f);
      zi_[t][j] = (yr[j] * kf.y + yi[j] * kf.x) * (1.f / 4096.f);
    }
  }
  __syncthreads();                          // all waves done reading sA
#pragma unroll
  for (int t = 0; t < 2; ++t) {
    int tile = wv * 2 + t, tm = (tile >> 2) * 16, tn = (tile & 3) * 16;
    int colg = tn + r15, rbase = tm + rbofs;
#pragma unroll
    for (int j = 0; j < 8; ++j) {
      int k = colg * 64 + rbase + j;        // Mz stored at natural index
      sA[k * 2 + 0] = zr_[t][j];
      sA[k * 2 + 1] = zi_[t][j];
    }
  }
  __syncthreads();

  // ========== inverse stage 1:  Ainv = conj(W64) @ Mz (4 GEMMs) ============
  v8f g1r[2], g1i[2];
#pragma unroll
  for (int t = 0; t < 2; ++t) {
    int tile = wv * 2 + t, tm = (tile >> 2) * 16, tn = (tile & 3) * 16;
    v8f cr = {}, ci = {};
#pragma unroll
    for (int k0 = 0; k0 < 64; k0 += 4) {
      v2f wr, wi; ldA_W(tm, k0, -1.f, wr, wi);
      v2f br = ldB_c(0, k0, tn);
      v2f bi = ldB_c(1, k0, tn);
      cr = wmma4(wr, br, cr);
      ci = wmma4(wr, bi, ci);
      ci = wmma4(wi, br, ci);
      v2f nbi; nbi.x = -bi.x; nbi.y = -bi.y;
      cr = wmma4(wi, nbi, cr);
    }
    g1r[t] = cr; g1i[t] = ci;
  }
  __syncthreads();
#pragma unroll
  for (int t = 0; t < 2; ++t) {
    int tile = wv * 2 + t, tm = (tile >> 2) * 16, tn = (tile & 3) * 16;
    int colg = tn + r15, rbase = tm + rbofs;
#pragma unroll
    for (int j = 0; j < 8; ++j) {
      int rowg = rbase + j;
      sA[(rowg * 64 + colg) * 2 + 0] = g1r[t][j];
      sA[(rowg * 64 + colg) * 2 + 1] = g1i[t][j];
    }
  }
  __syncthreads();

  // inverse twiddle: Ainv[c][b] *= exp(+2*pi*i*c*b/4096)
#pragma unroll
  for (int e = 0; e < 16; ++e) {
    int idx = tid * 16 + e;
    int c_ = idx >> 6, b_ = idx & 63;
    float s, c;
    __sincosf(2.f * PI_F * (float)(c_ * b_) * (1.f / 4096.f), &s, &c);
    float re = sA[idx * 2], im = sA[idx * 2 + 1];
    sA[idx * 2]     = re * c - im * s;
    sA[idx * 2 + 1] = re * s + im * c;
  }
  __syncthreads();

  // ========== inverse stage 2 (real part only): y = Re(Ainv @ conj(W64)) ===
#pragma unroll
  for (int t = 0; t < 2; ++t) {
    int tile = wv * 2 + t, tm = (tile >> 2) * 16, tn = (tile & 3) * 16;
    v8f rr = {};
#pragma unroll
    for (int k0 = 0; k0 < 64; k0 += 4) {
      v2f ar = ldA_c(0, tm, k0);
      v2f ai = ldA_c(1, tm, k0);
      v2f br, bi; ldB_W(k0, tn, -1.f, br, bi);   // conj(W64)
      rr = wmma4(ar, br, rr);
      v2f nbi; nbi.x = -bi.x; nbi.y = -bi.y;
      rr = wmma4(ai, nbi, rr);                   // Re = ar*br - ai*bi
    }
    int colg = tn + r15, rbase = tm + rbofs;
#pragma unroll
    for (int j = 0; j < 8; ++j) {
      int tt = colg * 64 + (rbase + j);          // time index t = e*64 + c
      yrow[tt] = rr[j];
    }
  }
}

// ---------------------------------------------------------------------------
// Kernel 5: transpose back (B,D,L) -> (B,L,D), fused  + D*x  epilogue
// ---------------------------------------------------------------------------
__global__ __launch_bounds__(256) void transpose_out(
    const float* __restrict__ yt, const float* __restrict__ x,
    const float* __restrict__ Dp, float* __restrict__ y)
{
  __shared__ float tile[32][33];
  int b = blockIdx.z, l0 = blockIdx.x * 32, d0 = blockIdx.y * 32;
  int tx = threadIdx.x, ty = threadIdx.y;
#pragma unroll
  for (int r = 0; r < 4; ++r) {
    int d = d0 + ty + r * 8;
    tile[ty + r * 8][tx] = yt[((size_t)b * DDIM + d) * SEQ + l0 + tx];
  }
  __syncthreads();
#pragma unroll
  for (int r = 0; r < 4; ++r) {
    int l = l0 + ty + r * 8;
    int d = d0 + tx;
    size_t o = ((size_t)b * SEQ + l) * DDIM + d;
    y[o] = tile[tx][ty + r * 8] + Dp[d] * x[o];
  }
}

// ---------------------------------------------------------------------------
// Launch. Inputs (setup_inputs order): x, lam_ri, P_ri, B_ri, C_ri, D, log_delta
// Workspace: Kf (8MB) + xT (32MB) + yT (32MB) = 72MB.
// ---------------------------------------------------------------------------
extern "C" void kernel_launch(void* const* d_in, const int* in_sizes, int n_in,
                              void* d_out, int out_size, void* d_ws, size_t ws_size,
                              hipStream_t stream)
{
  (void)in_sizes; (void)n_in; (void)out_size; (void)ws_size;
  const float*  x   = (const float*) d_in[0];
  const float2* lam = (const float2*)d_in[1];
  const float2* P   = (const float2*)d_in[2];
  const float2* Bc  = (const float2*)d_in[3];
  const float2* Cc  = (const float2*)d_in[4];
  const float*  Dp  = (const float*) d_in[5];
  const float*  ld  = (const float*) d_in[6];
  float* y = (float*)d_out;

  char*   ws = (char*)d_ws;
  float2* kf = (float2*)ws;                                        // D*L c64
  float*  xt = (float*)(ws + (size_t)DDIM * SEQ * sizeof(float2)); // B*D*L f32
  float*  yt = xt + (size_t)BATCH * DDIM * SEQ;                    // B*D*L f32

  cauchy_kernel<<<DDIM * SEQ / 256, 256, 0, stream>>>(lam, P, Bc, Cc, ld, kf);
  symm_kernel<<<(DDIM * 2049 + 255) / 256, 256, 0, stream>>>(kf);

  dim3 tb(32, 8);
  dim3 tg(SEQ / 32, DDIM / 32, BATCH);
  transpose_in<<<tg, tb, 0, stream>>>(x, xt);

  fftconv_kernel<<<BATCH * DDIM, 256, 0, stream>>>(xt, kf, yt);

  transpose_out<<<tg, tb, 0, stream>>>(yt, x, Dp, y);
}